// BSplineLayer_66425964200352
// MI455X (gfx1250) — compile-verified
//
#include <hip/hip_runtime.h>
#include <hip/hip_bf16.h>
#include <stdint.h>

// Piecewise-linear spline eval: u[4096,64,256] f32, knots/coefs[256,64] f32.
// Memory-bound (512 MiB streamed; ~23us HBM floor at 23.3 TB/s). No matmul
// structure -> no WMMA. CDNA5 paths used: TDM tensor_load_to_lds (table
// staging w/ hardware LDS padding), s_wait_tensorcnt, global_prefetch_b8,
// ds_load binary search.

#define M_CH      256   // channels (fastest dim of u)
#define NKNOT     64    // knots per channel
#define ROW_PAD   65    // 64 + 1 pad float per channel row (LDS bank spread)
#define BLOCK_T   256   // one thread per channel
#define ROWS_ITER 4     // rows of 256 channels per loop iteration

typedef unsigned int u32x4 __attribute__((ext_vector_type(4)));
typedef int          i32x8 __attribute__((ext_vector_type(8)));
typedef int          i32x4 __attribute__((ext_vector_type(4)));

// Issue one TDM descriptor: 2D tile (64 x 256 f32) global->LDS with padding
// of 1 DWORD after every 64 DWORDs => LDS rows land at stride 65 floats.
__device__ __forceinline__ void tdm_load_table(const float* gsrc, uint32_t lds_byte)
{
    const uint64_t ga = (uint64_t)(uintptr_t)gsrc;
    u32x4 g0;
    g0[0] = 1u;                                            // count=1 (valid user D#)
    g0[1] = lds_byte;                                      // lds_addr
    g0[2] = (uint32_t)ga;                                  // global_addr[31:0]
    g0[3] = ((uint32_t)(ga >> 32) & 0x01FFFFFFu)           // global_addr[56:32]
          | (2u << 30);                                    // type=2 ("image")
    i32x8 g1;
    g1[0] = (int)((2u << 16)        // data_size = 4 bytes
                | (1u << 20)        // pad_enable
                | (5u << 22)        // pad_interval: 64 DWORDs
                | (0u << 25));      // pad_amount: 1 DWORD
    g1[1] = (int)((uint32_t)NKNOT << 16);   // tensor_dim0[15:0]  (bits 63:48)
    g1[2] = (int)((uint32_t)M_CH  << 16);   // tensor_dim0 hi=0 | tensor_dim1 lo
    g1[3] = (int)((uint32_t)NKNOT << 16);   // tensor_dim1 hi=0 | tile_dim0=64
    g1[4] = (int)M_CH;                      // tile_dim1=256 | tile_dim2=0
    g1[5] = NKNOT;                          // tensor_dim0_stride[31:0] = 64
    g1[6] = 0;                              // stride0 hi | tensor_dim1_stride lo
    g1[7] = 0;                              // tensor_dim1_stride hi
    i32x4 gz4 = {0, 0, 0, 0};               // groups 2/3 unused (2D tensor)
    i32x8 gz8 = {0, 0, 0, 0, 0, 0, 0, 0};   // group 4 unused (clang-23 6-arg form)
    __builtin_amdgcn_tensor_load_to_lds(g0, g1, gz4, gz4, gz8, 0);
}

// Branchless lower_bound rank over 64 sorted knots (searchsorted side='left'),
// fused with the reference's clip(idx-1, 0, K-2). Probes depth 32..1: exact
// for ranks 0..63, and ranks 63/64 both clamp to interval 62.
__device__ __forceinline__ float spline_eval(const float* __restrict__ kc,
                                             const float* __restrict__ cc,
                                             float x)
{
    int idx = 0;
#pragma unroll
    for (int s = 32; s > 0; s >>= 1) {
        idx += (kc[idx + s - 1] < x) ? s : 0;
    }
    int i0 = idx - 1;
    i0 = i0 < 0 ? 0 : i0;
    i0 = i0 > (NKNOT - 2) ? (NKNOT - 2) : i0;
    const float k0 = kc[i0];
    const float k1 = kc[i0 + 1];
    const float c0 = cc[i0];
    const float c1 = cc[i0 + 1];
    const float t  = __fdividef(x - k0, (k1 - k0) + 1e-6f);  // v_rcp path; eps keeps denom safe
    return __builtin_fmaf(t, c1 - c0, c0);
}

__global__ void __launch_bounds__(BLOCK_T)
bspline_eval_kernel(const float* __restrict__ u,
                    const float* __restrict__ knots,
                    const float* __restrict__ coefs,
                    float* __restrict__ out,
                    unsigned int nrows)
{
    extern __shared__ float smem[];
    float* kn = smem;                        // [256][65] padded knots
    float* cf = smem + M_CH * ROW_PAD;       // [256][65] padded coefs

    // Wave 0 (uniform branch) issues both DMA descriptors; TDM ignores EXEC.
    if (threadIdx.x < 32) {
        tdm_load_table(knots, (uint32_t)(uintptr_t)kn);
        tdm_load_table(coefs, (uint32_t)(uintptr_t)cf);
        __builtin_amdgcn_s_wait_tensorcnt(0);
    }
    __syncthreads();

    const unsigned int c = threadIdx.x;      // this thread's channel
    const float* kc = kn + c * ROW_PAD;
    const float* cc = cf + c * ROW_PAD;

    const unsigned int ngroups = nrows >> 2;                     // groups of 4 rows
    const unsigned int gstride = gridDim.x;
    const unsigned int estep   = gstride * (ROWS_ITER * M_CH);   // elems per grid sweep

    for (unsigned int g = blockIdx.x; g < ngroups; g += gstride) {
        const unsigned int base = g * (ROWS_ITER * M_CH) + c;
        __builtin_prefetch(&u[base + estep], 0, 0);              // global_prefetch_b8

        const float x0 = __builtin_nontemporal_load(&u[base + 0 * M_CH]);
        const float x1 = __builtin_nontemporal_load(&u[base + 1 * M_CH]);
        const float x2 = __builtin_nontemporal_load(&u[base + 2 * M_CH]);
        const float x3 = __builtin_nontemporal_load(&u[base + 3 * M_CH]);

        const float r0 = spline_eval(kc, cc, x0);
        const float r1 = spline_eval(kc, cc, x1);
        const float r2 = spline_eval(kc, cc, x2);
        const float r3 = spline_eval(kc, cc, x3);

        __builtin_nontemporal_store(r0, &out[base + 0 * M_CH]);
        __builtin_nontemporal_store(r1, &out[base + 1 * M_CH]);
        __builtin_nontemporal_store(r2, &out[base + 2 * M_CH]);
        __builtin_nontemporal_store(r3, &out[base + 3 * M_CH]);
    }

    // Tail rows (nrows not a multiple of 4).
    for (unsigned int r = (ngroups << 2) + blockIdx.x; r < nrows; r += gstride) {
        const unsigned int base = r * M_CH + c;
        const float x = __builtin_nontemporal_load(&u[base]);
        __builtin_nontemporal_store(spline_eval(kc, cc, x), &out[base]);
    }
}

extern "C" void kernel_launch(void* const* d_in, const int* in_sizes, int n_in,
                              void* d_out, int out_size, void* d_ws, size_t ws_size,
                              hipStream_t stream)
{
    (void)n_in; (void)d_ws; (void)ws_size; (void)out_size;
    const float* u     = (const float*)d_in[0];
    const float* knots = (const float*)d_in[1];
    const float* coefs = (const float*)d_in[2];
    float*       out   = (float*)d_out;

    const unsigned int n     = (unsigned int)in_sizes[0];
    const unsigned int nrows = n / M_CH;     // 262144 for the reference shapes

    const size_t shmem = (size_t)2 * M_CH * ROW_PAD * sizeof(float);  // 133,120 B

    unsigned int grid = 4096;
    unsigned int maxg = (nrows + ROWS_ITER - 1) / ROWS_ITER;
    if (maxg == 0) maxg = 1;
    if (grid > maxg) grid = maxg;

    // >64 KiB dynamic LDS opt-in (gfx1250 WGP has 320 KiB).
    (void)hipFuncSetAttribute((const void*)bspline_eval_kernel,
                              hipFuncAttributeMaxDynamicSharedMemorySize,
                              (int)shmem);

    bspline_eval_kernel<<<dim3(grid), dim3(BLOCK_T), shmem, stream>>>(
        u, knots, coefs, out, nrows);
}